// DynamicConvolutionKernel_47785806135914
// MI455X (gfx1250) — compile-verified
//
#include <hip/hip_runtime.h>
#include <math.h>

// Problem constants (fixed by the reference setup)
#define B_   8
#define N_   4096
#define C_   1024
#define H_   256     // C/4
#define CK_  3072    // C*K, K=3

typedef __attribute__((ext_vector_type(2))) float v2f;
typedef __attribute__((ext_vector_type(8))) float v8f;

// ---------------------------------------------------------------------------
// Pass 1a: deterministic partial sums of x over N-chunks of 128 rows.
// grid (C/256, 32, B), block 256. part[b][chunk][c]
// ---------------------------------------------------------------------------
__global__ void __launch_bounds__(256)
ctx_partial(const float* __restrict__ x, float* __restrict__ part) {
  const int c     = blockIdx.x * 256 + threadIdx.x;
  const int chunk = blockIdx.y;
  const int b     = blockIdx.z;
  const int n0    = chunk * 128;
  const float* p  = x + ((size_t)b * N_ + n0) * C_ + c;
  float s = 0.f;
#pragma unroll 4
  for (int i = 0; i < 128; ++i) s += p[(size_t)i * C_];
  part[((size_t)b * 32 + chunk) * C_ + c] = s;
}

// Pass 1b: fold 32 partials -> ctx[b][c] = mean over N (deterministic order).
__global__ void __launch_bounds__(256)
ctx_finish(const float* __restrict__ part, float* __restrict__ ctx) {
  const int i = blockIdx.x * 256 + threadIdx.x;   // 8192 total
  const int b = i >> 10, c = i & (C_ - 1);
  float s = 0.f;
#pragma unroll
  for (int j = 0; j < 32; ++j) s += part[((size_t)b * 32 + j) * C_ + c];
  ctx[i] = s * (1.0f / (float)N_);
}

// ---------------------------------------------------------------------------
// Pass 2: kernel-generating MLP entirely with V_WMMA_F32_16X16X4_F32.
// One workgroup, 16 waves (wave32). M padded 8->16 (mask pad rows to 0).
//
// Lane layouts per CDNA5 ISA 7.12.2 (f32, 16x16x4):
//   A (16x4, 2 VGPR):  lane l -> row (l&15); K = (l>>4)*2 in .x, +1 in .y
//   B (4x16, 2 VGPR):  lane l -> col (l&15); K = (l>>4)*2 in .x, +1 in .y
//   C/D (16x16, v8f):  elem r of lane l -> row r+(l>>4)*8, col (l&15)
// ---------------------------------------------------------------------------
__device__ __forceinline__ float gelu_erf(float v) {
  return 0.5f * v * (1.0f + erff(v * 0.70710678118654752f));
}

__global__ void __launch_bounds__(512)
mlp_wmma(const float* __restrict__ ctx, const float* __restrict__ W1,
         const float* __restrict__ b1,  const float* __restrict__ W2,
         const float* __restrict__ b2,  float* __restrict__ kern) {
  __shared__ float h[16 * H_];                       // 16 KB of the 320 KB LDS
  const int tid  = threadIdx.x;
  const int w    = tid >> 5;                         // wave id 0..15
  const int lane = tid & 31;
  const int mn   = lane & 15;                        // A-row / B,C,D-column
  const int half = lane >> 4;
  const int koff = half * 2;
  const float rowmask = (mn < 8) ? 1.0f : 0.0f;      // pad rows 8..15 -> 0

  // ---- GEMM1: h16[16,256] = gelu(ctxP[16,1024] @ W1 + b1). Wave w owns cols [16w,16w+16)
  {
    const int n = w * 16 + mn;
    v8f acc = {};
    const float* ctxrow = ctx + (size_t)(mn & 7) * C_;
    for (int k = 0; k < C_; k += 4) {
      v2f a, bb;
      a.x  = ctxrow[k + koff]     * rowmask;
      a.y  = ctxrow[k + koff + 1] * rowmask;
      bb.x = W1[(size_t)(k + koff)     * H_ + n];
      bb.y = W1[(size_t)(k + koff + 1) * H_ + n];
      acc = __builtin_amdgcn_wmma_f32_16x16x4_f32(false, a, false, bb,
                                                  (short)0, acc, false, false);
    }
    const float bias = b1[n];
#pragma unroll
    for (int r = 0; r < 8; ++r)
      h[(size_t)(r + half * 8) * H_ + n] = gelu_erf(acc[r] + bias);
  }
  __syncthreads();

  // ---- GEMM2: kernels[16,3072] = h16 @ W2 + b2; 192 N-tiles, 12 per wave.
  for (int i = 0; i < 12; ++i) {
    const int t = w + i * 16;
    const int n = t * 16 + mn;
    v8f acc = {};
    for (int k = 0; k < H_; k += 4) {
      v2f a, bb;
      a.x  = h[(size_t)mn * H_ + k + koff];
      a.y  = h[(size_t)mn * H_ + k + koff + 1];
      bb.x = W2[(size_t)(k + koff)     * CK_ + n];
      bb.y = W2[(size_t)(k + koff + 1) * CK_ + n];
      acc = __builtin_amdgcn_wmma_f32_16x16x4_f32(false, a, false, bb,
                                                  (short)0, acc, false, false);
    }
    const float bias = b2[n];
    if (half == 0) {                                  // rows 0..7 are the real batch
#pragma unroll
      for (int r = 0; r < 8; ++r)
        kern[(size_t)r * CK_ + n] = acc[r] + bias;
    }
  }
}

// ---------------------------------------------------------------------------
// Pass 3: fused depthwise dynamic conv (K=3, same pad) + residual + LayerNorm.
// One block per (b, n) row; 256 threads x float4 = 1024 channels.
// ---------------------------------------------------------------------------
__global__ void __launch_bounds__(256)
conv_ln(const float* __restrict__ x, const float* __restrict__ kern,
        const float* __restrict__ gamma, const float* __restrict__ beta,
        float* __restrict__ out) {
  const int n   = blockIdx.x;
  const int b   = blockIdx.y;
  const int tid = threadIdx.x;
  const size_t rowoff = ((size_t)b * N_ + n) * C_;

  const float4 zero4 = {0.f, 0.f, 0.f, 0.f};
  const float4 xc = ((const float4*)(x + rowoff))[tid];
  const float4 xm = (n > 0)      ? ((const float4*)(x + rowoff - C_))[tid] : zero4;
  const float4 xp = (n < N_ - 1) ? ((const float4*)(x + rowoff + C_))[tid] : zero4;

  // 12 consecutive coefficients = kernels for channels 4*tid..4*tid+3 (c*3+j flat)
  const float4* kp = (const float4*)(kern + (size_t)b * CK_ + (size_t)tid * 12);
  const float4 k0 = kp[0], k1 = kp[1], k2 = kp[2];

  const float y0 = fmaf(k0.x, xm.x, fmaf(k0.y, xc.x, fmaf(k0.z, xp.x, xc.x)));
  const float y1 = fmaf(k0.w, xm.y, fmaf(k1.x, xc.y, fmaf(k1.y, xp.y, xc.y)));
  const float y2 = fmaf(k1.z, xm.z, fmaf(k1.w, xc.z, fmaf(k2.x, xp.z, xc.z)));
  const float y3 = fmaf(k2.y, xm.w, fmaf(k2.z, xc.w, fmaf(k2.w, xp.w, xc.w)));

  __shared__ float ssum[256];
  __shared__ float ssq[256];
  ssum[tid] = y0 + y1 + y2 + y3;
  ssq[tid]  = y0 * y0 + y1 * y1 + y2 * y2 + y3 * y3;
  __syncthreads();
#pragma unroll
  for (int s = 128; s > 0; s >>= 1) {
    if (tid < s) { ssum[tid] += ssum[tid + s]; ssq[tid] += ssq[tid + s]; }
    __syncthreads();
  }
  const float mu   = ssum[0] * (1.0f / (float)C_);
  const float var  = ssq[0]  * (1.0f / (float)C_) - mu * mu;
  const float rstd = rsqrtf(var + 1e-5f);

  const float4 g  = ((const float4*)gamma)[tid];
  const float4 be = ((const float4*)beta)[tid];
  float4 o;
  o.x = g.x * (y0 - mu) * rstd + be.x;
  o.y = g.y * (y1 - mu) * rstd + be.y;
  o.z = g.z * (y2 - mu) * rstd + be.z;
  o.w = g.w * (y3 - mu) * rstd + be.w;
  ((float4*)(out + rowoff))[tid] = o;
}

// ---------------------------------------------------------------------------
extern "C" void kernel_launch(void* const* d_in, const int* in_sizes, int n_in,
                              void* d_out, int out_size, void* d_ws, size_t ws_size,
                              hipStream_t stream) {
  const float* x     = (const float*)d_in[0];
  const float* W1    = (const float*)d_in[1];
  const float* b1    = (const float*)d_in[2];
  const float* W2    = (const float*)d_in[3];
  const float* b2    = (const float*)d_in[4];
  const float* gamma = (const float*)d_in[5];
  const float* beta  = (const float*)d_in[6];
  float* out = (float*)d_out;

  // workspace layout (floats): ctx[8192] | kern[24576] | part[8*32*1024]
  float* ctx  = (float*)d_ws;
  float* kern = ctx + (size_t)B_ * C_;
  float* part = kern + (size_t)B_ * CK_;

  ctx_partial<<<dim3(C_ / 256, 32, B_), 256, 0, stream>>>(x, part);
  ctx_finish <<<dim3((B_ * C_) / 256),  256, 0, stream>>>(part, ctx);
  mlp_wmma   <<<dim3(1), 512, 0, stream>>>(ctx, W1, b1, W2, b2, kern);
  conv_ln    <<<dim3(N_, B_), 256, 0, stream>>>(x, kern, gamma, beta, out);
}